// LRF_11330123727115
// MI455X (gfx1250) — compile-verified
//
#include <hip/hip_runtime.h>
#include <stdint.h>

// LRF kernel for MI455X / gfx1250. B=64, G=2048, S=64, C=3.
// One wave32 handles FOUR (b,g) groups (8 lanes per group, 8 points per lane):
// amortizes the wave-uniform eigen-solve 4x and shortens reductions to 3 shfl steps,
// bringing VALU time under the ~9us HBM roofline (206 MB @ 23.3 TB/s).
// CDNA5 paths: global_load_async_to_lds (ASYNCcnt) staging of 3 KB/wave tiles,
// V_WMMA_F32_16X16X4_F32 for the rotation einsum (rot^T = L^T * P^T per 16-pt chunk).

typedef __attribute__((ext_vector_type(2))) float v2f;
typedef __attribute__((ext_vector_type(8))) float v8f;

#define WAVES_PER_BLOCK 8
#define GROUPS_PER_WAVE 4
#define NGROUPS (64 * 2048)

struct F3 { float x, y, z; };
__device__ __forceinline__ F3 f3(float a, float b, float c) { F3 r; r.x=a; r.y=b; r.z=c; return r; }
__device__ __forceinline__ F3 crossf(F3 a, F3 b) {
    return f3(a.y*b.z - a.z*b.y, a.z*b.x - a.x*b.z, a.x*b.y - a.y*b.x);
}
__device__ __forceinline__ float dotf(F3 a, F3 b) { return a.x*b.x + a.y*b.y + a.z*b.z; }

// Reductions across the 8-lane subgroup that owns one group (xor masks 1,2,4 stay in-block).
__device__ __forceinline__ float red_sum8(float v) {
    v += __shfl_xor(v, 1, 32);
    v += __shfl_xor(v, 2, 32);
    v += __shfl_xor(v, 4, 32);
    return v;
}
__device__ __forceinline__ float red_max8(float v) {
    v = fmaxf(v, __shfl_xor(v, 1, 32));
    v = fmaxf(v, __shfl_xor(v, 2, 32));
    v = fmaxf(v, __shfl_xor(v, 4, 32));
    return v;
}

// Unit eigenvector of symmetric A for eigenvalue lam: best cross-product of rows of (A - lam I).
__device__ __forceinline__ F3 sym_eigvec(float a00, float a01, float a02,
                                         float a11, float a12, float a22, float lam) {
    F3 r0 = f3(a00 - lam, a01, a02);
    F3 r1 = f3(a01, a11 - lam, a12);
    F3 r2 = f3(a02, a12, a22 - lam);
    F3 c0 = crossf(r0, r1);
    F3 c1 = crossf(r0, r2);
    F3 c2 = crossf(r1, r2);
    float d0 = dotf(c0, c0), d1 = dotf(c1, c1), d2 = dotf(c2, c2);
    F3 best = c0; float db = d0;
    if (d1 > db) { best = c1; db = d1; }
    if (d2 > db) { best = c2; db = d2; }
    float inv = rsqrtf(db + 1e-30f);
    return f3(best.x * inv, best.y * inv, best.z * inv);
}

__global__ __launch_bounds__(256) void lrf_kernel(const float* __restrict__ neighbor,
                                                  float* __restrict__ rot_out,
                                                  float* __restrict__ lrf_out) {
    __shared__ __align__(16) float smem[WAVES_PER_BLOCK * GROUPS_PER_WAVE * 192]; // 3 KB/wave
    const int lane  = threadIdx.x & 31;
    const int wave  = threadIdx.x >> 5;
    const int g     = lane >> 3;          // subgroup 0..3 (which group this lane works on)
    const int t     = lane & 7;           // lane within subgroup; owns points 8t..8t+7
    const int gbase = (blockIdx.x * WAVES_PER_BLOCK + wave) * GROUPS_PER_WAVE;

    float* myl = &smem[wave * (GROUPS_PER_WAVE * 192)];

    // ---- Async-stage 3072 B (4 groups) to LDS: 6x global_load_async_to_lds_b128 ----
    {
        uint32_t lds0 = (uint32_t)(uintptr_t)myl; // low 32 bits of flat LDS addr = LDS offset
        uint64_t gsrc = (uint64_t)(uintptr_t)(neighbor + (size_t)gbase * 192);
#pragma unroll
        for (int k = 0; k < 6; ++k) {
            uint32_t la = lds0 + (uint32_t)(k * 512 + lane * 16);
            uint64_t ga = gsrc + (uint32_t)(k * 512 + lane * 16);
            asm volatile("global_load_async_to_lds_b128 %0, %1, off" :: "v"(la), "v"(ga) : "memory");
        }
        asm volatile("s_wait_asynccnt 0" ::: "memory");
    }

    // ---- Load this lane's 8 points (96 B contiguous, 16B aligned) as 6x ds_load_b128 ----
    float pv[24];
    {
        const float4* own4 = (const float4*)(myl + g * 192 + t * 24);
#pragma unroll
        for (int k = 0; k < 6; ++k) {
            float4 q = own4[k];
            pv[4*k+0] = q.x; pv[4*k+1] = q.y; pv[4*k+2] = q.z; pv[4*k+3] = q.w;
        }
    }

    // ---- Norms, max, weights ----
    float nrm[8];
    float lmax = -1e30f;
#pragma unroll
    for (int i = 0; i < 8; ++i) {
        float x = pv[3*i], y = pv[3*i+1], z = pv[3*i+2];
        float n = sqrtf(x*x + y*y + z*z);
        nrm[i] = n;
        lmax = fmaxf(lmax, n);
    }
    float mx = red_max8(lmax);
    float lsum = 0.0f;
#pragma unroll
    for (int i = 0; i < 8; ++i) lsum += (mx - nrm[i]);
    float sw   = red_sum8(lsum);
    float winv = 1.0f / (sw + 1e-6f);

    // ---- Weighted covariance (100x scaling only scales eigenvalues; dropped) ----
    float s00 = 0.f, s01 = 0.f, s02 = 0.f, s11 = 0.f, s12 = 0.f, s22 = 0.f;
#pragma unroll
    for (int i = 0; i < 8; ++i) {
        float x = pv[3*i], y = pv[3*i+1], z = pv[3*i+2];
        float w = (mx - nrm[i]) * winv;
        float wx = w * x, wy = w * y, wz = w * z;
        s00 += wx * x; s01 += wx * y; s02 += wx * z;
        s11 += wy * y; s12 += wy * z; s22 += wz * z;
    }
    float a00 = red_sum8(s00), a01 = red_sum8(s01), a02 = red_sum8(s02);
    float a11 = red_sum8(s11), a12 = red_sum8(s12), a22 = red_sum8(s22);

    // ---- Analytic symmetric 3x3 eigenvalues (Smith); cos via v_cos_f32 ----
    float q   = (a00 + a11 + a22) * (1.0f / 3.0f);
    float pp1 = a01*a01 + a02*a02 + a12*a12;
    float b00 = a00 - q, b11 = a11 - q, b22 = a22 - q;
    float p2  = b00*b00 + b11*b11 + b22*b22 + 2.0f * pp1;
    float p   = sqrtf(p2 * (1.0f / 6.0f)) + 1e-30f;
    float ip  = 1.0f / p;
    float c00 = b00*ip, c01 = a01*ip, c02 = a02*ip;
    float c11 = b11*ip, c12 = a12*ip, c22 = b22*ip;
    float detB = c00*(c11*c22 - c12*c12) - c01*(c01*c22 - c12*c02) + c02*(c01*c12 - c11*c02);
    float r    = fminf(1.0f, fmaxf(-1.0f, 0.5f * detB));
    float phi  = acosf(r) * (1.0f / 3.0f);
    float e_big   = q + 2.0f * p * __cosf(phi);                  // largest eigenvalue
    float e_small = q + 2.0f * p * __cosf(phi + 2.09439510239f); // smallest eigenvalue

    F3 zax = sym_eigvec(a00, a01, a02, a11, a12, a22, e_small); // eigvec col 0
    F3 xax = sym_eigvec(a00, a01, a02, a11, a12, a22, e_big);   // eigvec col 2

    // ---- Sign disambiguation: flip if fewer than half (32) projections are > 0 ----
    float czl = 0.f, cxl = 0.f;
#pragma unroll
    for (int i = 0; i < 8; ++i) {
        F3 pt = f3(pv[3*i], pv[3*i+1], pv[3*i+2]);
        czl += (dotf(zax, pt) > 0.0f) ? 1.0f : 0.0f;
        cxl += (dotf(xax, pt) > 0.0f) ? 1.0f : 0.0f;
    }
    float cz = red_sum8(czl);
    float cx = red_sum8(cxl);
    float fz = (cz < 32.0f) ? -1.0f : 1.0f;
    float fx = (cx < 32.0f) ? -1.0f : 1.0f;
    zax = f3(zax.x * fz, zax.y * fz, zax.z * fz);
    xax = f3(xax.x * fx, xax.y * fx, xax.z * fx);
    F3 yax = crossf(zax, xax);

    // ---- lrfs[c][d] = component c of axis d (z,y,x): lanes t=0..2 store row c=t ----
    if (t < 3) {
        float rc0 = (t == 0) ? zax.x : (t == 1) ? zax.y : zax.z;
        float rc1 = (t == 0) ? yax.x : (t == 1) ? yax.y : yax.z;
        float rc2 = (t == 0) ? xax.x : (t == 1) ? xax.y : xax.z;
        float* o = lrf_out + (size_t)(gbase + g) * 9 + t * 3;
        o[0] = rc0; o[1] = rc1; o[2] = rc2;
    }

    // ---- Rotation via V_WMMA_F32_16X16X4_F32, one group at a time ----
    // rot^T(3x64) = L^T(3x4pad) * P^T(4x64) in four N=16 chunks.
    // A (16x4 f32): lanes 0-15: M=lane, VGPR0=K0 / VGPR1=K1; lanes 16-31: K2/K3.
    // A[m][k] = L[k][m] = component k of axis m (axes z,y,x for m=0,1,2; else 0).
    const int   m   = lane & 15;
    const float flo = (lane < 16) ? 1.0f : 0.0f; // arithmetic half-select (branchless)
#pragma unroll
    for (int g2 = 0; g2 < 4; ++g2) {
        const int src = g2 * 8; // broadcast L of group g2 from its subgroup's first lane
        float zbx = __shfl(zax.x, src, 32), zby = __shfl(zax.y, src, 32), zbz = __shfl(zax.z, src, 32);
        float ybx = __shfl(yax.x, src, 32), yby = __shfl(yax.y, src, 32), ybz = __shfl(yax.z, src, 32);
        float xbx = __shfl(xax.x, src, 32), xby = __shfl(xax.y, src, 32), xbz = __shfl(xax.z, src, 32);

        float ac0 = (m == 0) ? zbx : (m == 1) ? ybx : (m == 2) ? xbx : 0.0f; // component 0 of axis m
        float ac1 = (m == 0) ? zby : (m == 1) ? yby : (m == 2) ? xby : 0.0f; // component 1
        float ac2 = (m == 0) ? zbz : (m == 1) ? ybz : (m == 2) ? xbz : 0.0f; // component 2
        v2f A;
        A.x = ac2 + (ac0 - ac2) * flo; // lanes<16: K0 comp; lanes>=16: K2 comp
        A.y = ac1 * flo;               // lanes<16: K1 comp; lanes>=16: K3 = 0

        const float* gl   = myl + g2 * 192;
        float* rbase      = rot_out + (size_t)(gbase + g2) * 192;
#pragma unroll
        for (int chunk = 0; chunk < 4; ++chunk) {
            const int pt = chunk * 16 + m;
            float px = gl[3*pt + 0], py = gl[3*pt + 1], pz = gl[3*pt + 2];
            // B (4x16): lanes 0-15: N=lane, K0/K1; lanes 16-31: N=lane-16, K2/K3.
            v2f Bv;
            Bv.x = pz + (px - pz) * flo; // all loads used by all lanes -> no exec branching
            Bv.y = py * flo;
            v8f C = {};
            C = __builtin_amdgcn_wmma_f32_16x16x4_f32(false, A, false, Bv,
                                                      (short)0, C, false, false);
            // D: lanes 0-15, accumulator VGPR r = output dim r, N = point-in-chunk.
            if (lane < 16) {
                float* o = rbase + pt * 3;
                o[0] = C[0]; o[1] = C[1]; o[2] = C[2];
            }
        }
    }
}

extern "C" void kernel_launch(void* const* d_in, const int* in_sizes, int n_in,
                              void* d_out, int out_size, void* d_ws, size_t ws_size,
                              hipStream_t stream) {
    const float* neighbor = (const float*)d_in[0]; // (64,2048,64,3) f32
    // d_in[1] = center: unused by the reference computation.
    float* rot = (float*)d_out;                    // (64,2048,64,3)
    float* lrf = rot + (size_t)64 * 2048 * 64 * 3; // (64,2048,3,3)
    dim3 grid(NGROUPS / (WAVES_PER_BLOCK * GROUPS_PER_WAVE)); // 4096 blocks x 256 threads
    lrf_kernel<<<grid, dim3(256), 0, stream>>>(neighbor, rot, lrf);
}